// IntervalClusterTripletFT_51316269252889
// MI455X (gfx1250) — compile-verified
//
#include <hip/hip_runtime.h>
#include <math.h>

// Problem constants from reference: C=256 clusters, K=16, D=256 -> N=4096
#define N_TOT 4096
#define DIM   256
#define CK    16
#define NC    256
#define EPS_F 1e-6f

typedef float v2f __attribute__((ext_vector_type(2)));
typedef float v8f __attribute__((ext_vector_type(8)));

// ---------------- Kernel 1: squared norms --------------------------------
__global__ __launch_bounds__(256) void sqnorm_kernel(const float* __restrict__ E,
                                                     float* __restrict__ sq) {
  const int wave = threadIdx.x >> 5;
  const int lane = threadIdx.x & 31;
  const int row  = blockIdx.x * 8 + wave;          // 512 blocks * 8 waves = 4096 rows
  const float* p = E + (size_t)row * DIM;
  float s = 0.f;
  #pragma unroll
  for (int j = lane; j < DIM; j += 32) { float x = p[j]; s += x * x; }
  #pragma unroll
  for (int off = 16; off; off >>= 1) s += __shfl_xor(s, off, 32);
  if (lane == 0) sq[row] = s;
}

// ---------------- Kernel 2: fused dist + hardest pos/neg + loss terms ----
// One workgroup per cluster (16 rows). 8 waves, each wave computes 32 of the
// 256 column tiles of the 16x4096 Gram row-block with V_WMMA_F32_16X16X4_F32.
__global__ __launch_bounds__(256) void triplet_main(const float* __restrict__ E,
                                                    const float* __restrict__ sq,
                                                    float* __restrict__ contrib) {
  __shared__ float As[16][260];          // cluster rows, padded stride vs 64 banks
  __shared__ float sqA[16];
  __shared__ float Ld[8][16][17];        // per-wave 16x16 dist tile (padded)
  __shared__ float redNegV[128]; __shared__ int redNegI[128];
  __shared__ float redPosV[128]; __shared__ int redPosI[128];
  __shared__ int   finalPos[16]; __shared__ int finalNeg[16];

  const int c      = blockIdx.x;
  const int tid    = threadIdx.x;
  const int wave   = tid >> 5;
  const int lane   = tid & 31;
  const int half   = lane >> 4;          // wave32: lanes 0-15 / 16-31
  const int lane16 = lane & 15;

  // Stage the cluster's 16x256 A-block into LDS (coalesced float4 loads).
  {
    int r = tid >> 4, cseg = (tid & 15) * 16;
    const float4* src = (const float4*)(E + (size_t)(c * CK + r) * DIM + cseg);
    float4* dst = (float4*)&As[r][cseg];
    dst[0] = src[0]; dst[1] = src[1]; dst[2] = src[2]; dst[3] = src[3];
  }
  if (tid < 16) sqA[tid] = sq[c * CK + tid];
  __syncthreads();

  float negV = 3.0e38f; int negI = 0x7fffffff;   // hardest negative (min dist)
  float posV = -1.0f;   int posI = 0x7fffffff;   // hardest positive (max dist)

  for (int it = 0; it < 32; ++it) {
    const int t       = wave * 32 + it;          // column tile 0..255
    const int colBase = t * 16;

    // ---- 16x16 Gram tile: D = A(16x256) * B(256x16), B = E_tile^T ----
    // f32 WMMA frag layout (ISA 7.12.2): A 16x4 -> lane m=L%16, VGPR0/1 hold
    // K = 2*(L/16) + {0,1}; B mirrored (lane = column N, halves = K rows).
    v8f acc = {};
    const float* aRow = &As[lane16][2 * half];
    const float* bRow = E + (size_t)(colBase + lane16) * DIM + 2 * half;
    #pragma unroll 8
    for (int kb = 0; kb < 64; ++kb) {
      v2f a = *(const v2f*)(aRow + kb * 4);
      v2f b = *(const v2f*)(bRow + kb * 4);
      acc = __builtin_amdgcn_wmma_f32_16x16x4_f32(false, a, false, b,
                                                  (short)0, acc, false, false);
    }

    // dist = sqrt(max(||a||^2 + ||b||^2 - 2ab, 0)); C layout: VGPR v <-> row
    // M = v + 8*half, col N = lane16. Stage transposed into the wave's tile.
    const float sqn = sq[colBase + lane16];
    #pragma unroll
    for (int v = 0; v < 8; ++v) {
      const int m = v + 8 * half;
      const float d2 = sqA[m] + sqn - 2.0f * acc[v];
      Ld[wave][m][lane16] = sqrtf(fmaxf(d2, 0.0f));
    }
    __syncthreads();   // uniform trip count across all 8 waves

    // Row-per-lane scan: lane owns row lane16, scans 8 cols (half*8 + j).
    const bool isPos = (t == c);   // diagonal tile = same-cluster columns
    #pragma unroll
    for (int j = 0; j < 8; ++j) {
      const int n = 8 * half + j;
      const float v = Ld[wave][lane16][n];
      const int g = colBase + n;
      if (isPos) {
        if (v > posV || (v == posV && g < posI)) { posV = v; posI = g; }
      } else {
        if (v < negV || (v == negV && g < negI)) { negV = v; negI = g; }
      }
    }
    __syncthreads();
  }

  // Combine column-halves within the wave (wave32 shuffle).
  {
    float ov = __shfl_xor(negV, 16, 32); int oi = __shfl_xor(negI, 16, 32);
    if (ov < negV || (ov == negV && oi < negI)) { negV = ov; negI = oi; }
    ov = __shfl_xor(posV, 16, 32); oi = __shfl_xor(posI, 16, 32);
    if (ov > posV || (ov == posV && oi < posI)) { posV = ov; posI = oi; }
  }
  if (half == 0) {
    redNegV[wave * 16 + lane16] = negV; redNegI[wave * 16 + lane16] = negI;
    redPosV[wave * 16 + lane16] = posV; redPosI[wave * 16 + lane16] = posI;
  }
  __syncthreads();

  // Combine across the 8 waves (fixed order -> deterministic tie-breaks).
  if (tid < 16) {
    float nv = redNegV[tid]; int ni = redNegI[tid];
    float pv = redPosV[tid]; int pi = redPosI[tid];
    for (int w = 1; w < 8; ++w) {
      float v2 = redNegV[w * 16 + tid]; int i2 = redNegI[w * 16 + tid];
      if (v2 < nv || (v2 == nv && i2 < ni)) { nv = v2; ni = i2; }
      v2 = redPosV[w * 16 + tid]; i2 = redPosI[w * 16 + tid];
      if (v2 > pv || (v2 == pv && i2 < pi)) { pv = v2; pi = i2; }
    }
    finalNeg[tid] = ni; finalPos[tid] = pi;
  }
  __syncthreads();

  // Loss terms: d_ap/d_an recomputed with elementwise +EPS (reference semantics).
  // Wave w handles rows 2w, 2w+1; anchor & positive rows come from LDS.
  #pragma unroll
  for (int rr = 0; rr < 2; ++rr) {
    const int r = wave * 2 + rr;
    const int i = c * CK + r;
    const int p = finalPos[r] - c * CK;               // positive is in-cluster
    const float* eneg = E + (size_t)finalNeg[r] * DIM;
    float sAp = 0.f, sAn = 0.f;
    for (int j = lane; j < DIM; j += 32) {
      const float xi = As[r][j];
      const float dp = xi - As[p][j] + EPS_F;
      const float dn = xi - eneg[j]  + EPS_F;
      sAp += dp * dp; sAn += dn * dn;
    }
    #pragma unroll
    for (int off = 16; off; off >>= 1) {
      sAp += __shfl_xor(sAp, off, 32);
      sAn += __shfl_xor(sAn, off, 32);
    }
    if (lane == 0)
      contrib[i] = fmaxf(sqrtf(sAp) - sqrtf(sAn) + 1.0f, 0.0f);
  }
}

// ---------------- Kernel 3: deterministic mean ---------------------------
__global__ __launch_bounds__(256) void reduce_kernel(const float* __restrict__ contrib,
                                                     float* __restrict__ out) {
  __shared__ float s[256];
  float a = 0.f;
  for (int k = 0; k < 16; ++k) a += contrib[threadIdx.x + 256 * k];  // fixed order
  s[threadIdx.x] = a;
  __syncthreads();
  for (int off = 128; off; off >>= 1) {
    if (threadIdx.x < off) s[threadIdx.x] += s[threadIdx.x + off];
    __syncthreads();
  }
  if (threadIdx.x == 0) out[0] = s[0] * (1.0f / (float)N_TOT);
}

// ---------------- Launch --------------------------------------------------
extern "C" void kernel_launch(void* const* d_in, const int* in_sizes, int n_in,
                              void* d_out, int out_size, void* d_ws, size_t ws_size,
                              hipStream_t stream) {
  (void)in_sizes; (void)n_in; (void)out_size; (void)ws_size;
  const float* E = (const float*)d_in[0];       // [256,16,256] fp32 -> [4096,256]
  float* sq      = (float*)d_ws;                // 4096 floats
  float* contrib = sq + N_TOT;                  // 4096 floats

  sqnorm_kernel<<<dim3(N_TOT / 8), dim3(256), 0, stream>>>(E, sq);
  triplet_main <<<dim3(NC),        dim3(256), 0, stream>>>(E, sq, contrib);
  reduce_kernel<<<dim3(1),         dim3(256), 0, stream>>>(contrib, (float*)d_out);
}